// ChannelPruning_48155173322822
// MI455X (gfx1250) — compile-verified
//
#include <hip/hip_runtime.h>
#include <hip/hip_bf16.h>

typedef __attribute__((ext_vector_type(2))) float v2f;
typedef __attribute__((ext_vector_type(4))) float v4f;
typedef __attribute__((ext_vector_type(8))) float v8f;

#define B_DIM   64
#define C_IN    512
#define C_OUT   512
#define HW      784        // 28*28
#define K_PRUNE 256        // int(C_OUT * 0.5)

// ---------------------------------------------------------------------------
// Kernel 1: s[b,c] = mean(|x[b,c,:,:]|)   -- the bandwidth-bound phase.
// One wave32 per (b,c) channel; 784 contiguous floats = 196 float4 chunks.
// Non-temporal 128-bit loads: x is streamed exactly once (103 MB), no reuse.
// ---------------------------------------------------------------------------
__global__ void absmean_kernel(const float* __restrict__ x,
                               float* __restrict__ s) {
    const int gwave = (blockIdx.x * blockDim.x + threadIdx.x) >> 5; // (b,c) id
    const int lane  = threadIdx.x & 31;
    if (gwave >= B_DIM * C_IN) return;

    const v4f* p = (const v4f*)(x + (size_t)gwave * HW);
    float sum = 0.0f;
#pragma unroll
    for (int it = 0; it < 7; ++it) {            // ceil(196/32) = 7
        int i4 = it * 32 + lane;
        if (i4 < HW / 4) {
            v4f v = __builtin_nontemporal_load(&p[i4]);
            sum += fabsf(v.x) + fabsf(v.y) + fabsf(v.z) + fabsf(v.w);
        }
    }
    // wave32 reduction
#pragma unroll
    for (int off = 16; off > 0; off >>= 1)
        sum += __shfl_down(sum, off, 32);
    if (lane == 0)
        s[gwave] = sum * (1.0f / (float)HW);
}

// ---------------------------------------------------------------------------
// Kernel 2: g = relu(S @ W[:, :512]^T + (b + 0.5 * W[:,512]))  via f32 WMMA.
// One wave per 16x16 tile of g; M=64, N=512 -> 4*32 = 128 waves.
// V_WMMA_F32_16X16X4_F32 layouts (ISA 7.12.2):
//   A 16x4 : lane l<16 holds M=l, K={0,1} in vgpr{0,1}; l>=16 holds K={2,3}
//   B 4x16 : lane l<16 holds N=l, K={0,1} in vgpr{0,1}; l>=16 holds K={2,3}
//   C 16x16: vgpr i holds M = i + 8*(lane>=16), N = lane&15
// ---------------------------------------------------------------------------
__global__ void gate_gemm_kernel(const float* __restrict__ S,   // [64,512]
                                 const float* __restrict__ W,   // [512,513]
                                 const float* __restrict__ bias,// [512]
                                 float* __restrict__ G) {       // [64,512]
    const int lane   = threadIdx.x & 31;
    const int waveId = blockIdx.x * (blockDim.x >> 5) + (threadIdx.x >> 5);
    const int tileM  = waveId >> 5;          // 0..3
    const int tileN  = waveId & 31;          // 0..31
    const int half   = lane >> 4;            // selects K pair {0,1} vs {2,3}
    const int l16    = lane & 15;
    const int mRow   = tileM * 16 + l16;     // A row held by this lane
    const int nCol   = tileN * 16 + l16;     // B col held by this lane

    const float* __restrict__ Srow = S + (size_t)mRow * C_IN + 2 * half;
    const float* __restrict__ Wrow = W + (size_t)nCol * (C_IN + 1) + 2 * half;

    v8f acc = {};
    for (int k = 0; k < C_IN; k += 4) {
        v2f a, b;
        a.x = Srow[k + 0];
        a.y = Srow[k + 1];
        b.x = Wrow[k + 0];
        b.y = Wrow[k + 1];
        acc = __builtin_amdgcn_wmma_f32_16x16x4_f32(
            /*neg_a=*/false, a, /*neg_b=*/false, b,
            /*c_mod=*/(short)0, acc, /*reuse_a=*/false, /*reuse_b=*/false);
    }

    // effective bias folds the constant rate column: b + 0.5 * W[:,512]
    const float beff = bias[nCol] + 0.5f * W[(size_t)nCol * (C_IN + 1) + C_IN];
#pragma unroll
    for (int i = 0; i < 8; ++i) {
        int m  = tileM * 16 + i + half * 8;
        float g = acc[i] + beff;
        g = g > 0.0f ? g : 0.0f;               // relu
        G[(size_t)m * C_OUT + nCol] = g;
    }
}

// ---------------------------------------------------------------------------
// Kernel 3: zero the K smallest (stable, matching top_k tie-break by index),
// then t = t / sum(t) * C_OUT.  One 512-thread block per batch row.
// ---------------------------------------------------------------------------
__global__ void prune_kernel(const float* __restrict__ G,
                             float* __restrict__ out) {
    __shared__ float row[C_OUT];
    __shared__ float wsum[16];
    __shared__ float total;

    const int b = blockIdx.x;
    const int t = threadIdx.x;

    const float v = G[(size_t)b * C_OUT + t];
    row[t] = v;
    __syncthreads();

    // stable rank: # of elements that sort strictly before (value, index)
    int rank = 0;
    for (int j = 0; j < C_OUT; ++j) {
        float u = row[j];
        rank += (u < v) || (u == v && j < t);
    }
    const float kept = (rank >= K_PRUNE) ? v : 0.0f;

    // block-wide sum of kept values (16 waves of 32)
    float ssum = kept;
#pragma unroll
    for (int off = 16; off > 0; off >>= 1)
        ssum += __shfl_down(ssum, off, 32);
    if ((t & 31) == 0) wsum[t >> 5] = ssum;
    __syncthreads();
    if (t == 0) {
        float tt = 0.0f;
#pragma unroll
        for (int i = 0; i < 16; ++i) tt += wsum[i];
        total = tt;
    }
    __syncthreads();

    out[(size_t)b * C_OUT + t] = (kept / total) * (float)C_OUT;
}

// ---------------------------------------------------------------------------
extern "C" void kernel_launch(void* const* d_in, const int* in_sizes, int n_in,
                              void* d_out, int out_size, void* d_ws, size_t ws_size,
                              hipStream_t stream) {
    const float* x    = (const float*)d_in[0];   // [64,512,28,28]
    const float* W    = (const float*)d_in[1];   // [512,513]
    const float* bias = (const float*)d_in[2];   // [512]
    float* out = (float*)d_out;                  // [64,512] (x1x1)

    float* s = (float*)d_ws;                     // [64,512]
    float* g = s + B_DIM * C_IN;                 // [64,512]

    // Phase 1: 32768 waves, 8 waves (256 threads) per block
    {
        const int waves  = B_DIM * C_IN;
        const int block  = 256;
        const int blocks = (waves * 32) / block; // 4096
        absmean_kernel<<<blocks, block, 0, stream>>>(x, s);
    }
    // Phase 2: 128 waves -> 16 blocks x 256 threads
    gate_gemm_kernel<<<16, 256, 0, stream>>>(s, W, bias, g);

    // Phase 3: one block per batch row
    prune_kernel<<<B_DIM, C_OUT, 0, stream>>>(g, out);
}